// GradPooling_63196148793926
// MI455X (gfx1250) — compile-verified
//
#include <hip/hip_runtime.h>

// GradPooling for x:(32,224,224,64) f32, lamb scalar -> out:(32,112,112,64) f32.
// Bandwidth-bound (~0.55 GB @ 23.3 TB/s ~ 25us). Strategy: float4 (b128)
// coalesced loads, NT temporal hints on single-use streams, NT store on the
// output, gfx1250 global_prefetch for the next grid-stride iteration.

typedef float v4f __attribute__((ext_vector_type(4)));

#define XH 224
#define XW 224
#define XC 64
#define OH 112
#define OW 112
// 16 float4 chunks per pixel (64 channels)

__device__ __forceinline__ v4f vmax4(v4f a, v4f b) {
  v4f r;
  r.x = fmaxf(a.x, b.x); r.y = fmaxf(a.y, b.y);
  r.z = fmaxf(a.z, b.z); r.w = fmaxf(a.w, b.w);
  return r;
}

__device__ __forceinline__ v4f vabs4(v4f a) {
  v4f r;
  r.x = __builtin_fabsf(a.x); r.y = __builtin_fabsf(a.y);
  r.z = __builtin_fabsf(a.z); r.w = __builtin_fabsf(a.w);
  return r;
}

__global__ void __launch_bounds__(256) gradpool_kernel(
    const float* __restrict__ x, const float* __restrict__ lambp,
    float* __restrict__ out, int total)
{
  const float lam = lambp[0];
  const v4f zero = {0.f, 0.f, 0.f, 0.f};

  const int tid = blockIdx.x * blockDim.x + threadIdx.x;
  const int gstride = gridDim.x * blockDim.x;

  for (int n = tid; n < total; n += gstride) {
    // ---- prefetch next iteration's streaming (pooling) rows ----
    int n2 = n + gstride;
    if (n2 < total) {
      unsigned t2 = (unsigned)n2;
      int c42 = (int)(t2 & 15u); t2 >>= 4;
      int ow2 = (int)(t2 % OW); t2 /= OW;
      int oh2 = (int)(t2 % OH); int b2 = (int)(t2 / OH);
      const float* xb2 = x + ((size_t)b2 * XH * XW * XC) + (size_t)c42 * 4;
      int r12 = 2 * oh2, q12 = 2 * ow2;
      __builtin_prefetch(xb2 + ((size_t)r12 * XW + q12) * XC, 0, 0);
      if (r12 > 0)
        __builtin_prefetch(xb2 + ((size_t)(r12 - 1) * XW + q12) * XC, 0, 0);
    }

    // ---- decode (b, oh, ow, chunk) ----
    unsigned t = (unsigned)n;
    const int c4 = (int)(t & 15u); t >>= 4;
    const int ow = (int)(t % OW); t /= OW;
    const int oh = (int)(t % OH);
    const int b  = (int)(t / OH);

    const float* xb = x + ((size_t)b * XH * XW * XC) + (size_t)c4 * 4;

    // ---- stride-2 pooling window; OOB (padded) entries are literal 0 ----
    const int r0 = 2 * oh - 1, r1 = 2 * oh;
    const int q0 = 2 * ow - 1, q1 = 2 * ow;

    v4f v00 = (r0 >= 0 && q0 >= 0)
        ? __builtin_nontemporal_load((const v4f*)(xb + ((size_t)r0 * XW + q0) * XC)) : zero;
    v4f v01 = (r0 >= 0)
        ? __builtin_nontemporal_load((const v4f*)(xb + ((size_t)r0 * XW + q1) * XC)) : zero;
    v4f v10 = (q0 >= 0)
        ? __builtin_nontemporal_load((const v4f*)(xb + ((size_t)r1 * XW + q0) * XC)) : zero;
    v4f v11 =
          __builtin_nontemporal_load((const v4f*)(xb + ((size_t)r1 * XW + q1) * XC));

    v4f mx = vmax4(vmax4(v00, v01), vmax4(v10, v11));
    v4f mn = (v00 + v01 + v10 + v11) * 0.25f;

    // ---- stride-1 selector (top-left 113x113 region, re-read ~3x: keep RT) ----
    const v4f xc = *(const v4f*)(xb + ((size_t)(oh + 1) * XW + (ow + 1)) * XC);
    const v4f xu = *(const v4f*)(xb + ((size_t)(oh    ) * XW + (ow + 1)) * XC);
    const v4f xl = *(const v4f*)(xb + ((size_t)(oh + 1) * XW + (ow    )) * XC);

    v4f d = (vabs4(xc - xu) + vabs4(xc - xl)) * 2.0f;

    v4f res;
    res.x = (d.x > lam) ? mx.x : mn.x;
    res.y = (d.y > lam) ? mx.y : mn.y;
    res.z = (d.z > lam) ? mx.z : mn.z;
    res.w = (d.w > lam) ? mx.w : mn.w;

    __builtin_nontemporal_store(res, (v4f*)(out + (size_t)n * 4));
  }
}

extern "C" void kernel_launch(void* const* d_in, const int* in_sizes, int n_in,
                              void* d_out, int out_size, void* d_ws, size_t ws_size,
                              hipStream_t stream) {
  (void)in_sizes; (void)n_in; (void)d_ws; (void)ws_size;
  const float* x    = (const float*)d_in[0];
  const float* lamb = (const float*)d_in[1];
  float* out = (float*)d_out;

  const int total = out_size / 4;       // one float4 chunk per work-item
  dim3 block(256);                      // 8 wave32 waves per workgroup
  dim3 grid(6144);                      // persistent grid-stride, ~4-5 items/thread
  gradpool_kernel<<<grid, block, 0, stream>>>(x, lamb, out, total);
}